// GCN_70720931496421
// MI455X (gfx1250) — compile-verified
//
#include <hip/hip_runtime.h>
#include <math.h>

typedef __attribute__((ext_vector_type(2))) float v2f;
typedef __attribute__((ext_vector_type(8))) float v8f;

#define DIN 128
#define ROWTILE 64
#define XSTRIDE (DIN + 4)                      // 132 floats -> conflict-free b64 reads
#define LDSX_FLOATS (ROWTILE * XSTRIDE)        // 8448
#define LDSW_FLOATS (DIN * XSTRIDE)            // 16896
#define GEMM_LDS_BYTES ((LDSX_FLOATS + LDSW_FLOATS) * 4)   // 101376 B (< 320 KB WGP LDS)

// ---------------------------------------------------------------------------
// Degree / normalization
// ---------------------------------------------------------------------------
__global__ void degree_count(const int* __restrict__ dst, float* __restrict__ deg, int E) {
    int e = blockIdx.x * blockDim.x + threadIdx.x;
    if (e < E) atomicAdd(&deg[dst[e]], 1.0f);
}

__global__ void deg_to_dinv(float* __restrict__ deg, int n) {
    int i = blockIdx.x * blockDim.x + threadIdx.x;
    if (i < n) deg[i] = rsqrtf(deg[i] + 1.0f);
}

// ---------------------------------------------------------------------------
// FP32 WMMA GEMM: Y[row,col] = sum_k X[row,k] * W[k,col]   (K = 128 fixed)
// Block = 256 thr (8 waves) computes a 64-row x 128-col tile.
// Wave w owns cols [16w, 16w+16) and 4 row sub-tiles (4 accumulators).
// W staged in LDS transposed ([col][k], zero-padded for col >= n_cols):
// K-loop is branch-free -> ds_load_2addr_b64 + v_wmma_f32_16x16x4_f32 only.
// LDY templated so the epilogue is one address calc + immediate-offset stores.
// ---------------------------------------------------------------------------
template <int LDY>
__global__ void __launch_bounds__(256)
gemm_wmma(const float* __restrict__ X, const float* __restrict__ W,
          float* __restrict__ Y, int n_rows, int n_cols, int n_valid, int ldW) {
    extern __shared__ float lds[];
    float* ldsX = lds;                 // [ROWTILE][XSTRIDE]
    float* ldsW = lds + LDSX_FLOATS;   // [DIN][XSTRIDE]  indexed [col][k]

    const int tid  = threadIdx.x;
    const int row0 = blockIdx.x * ROWTILE;

    // Stage X tile (64 x 128) as float4, zero-pad rows >= n_rows
    for (int i = tid; i < ROWTILE * 32; i += 256) {
        int r = i >> 5, c4 = i & 31;
        float4 v = make_float4(0.f, 0.f, 0.f, 0.f);
        if (row0 + r < n_rows)
            v = reinterpret_cast<const float4*>(X + (size_t)(row0 + r) * DIN)[c4];
        *reinterpret_cast<float4*>(&ldsX[r * XSTRIDE + c4 * 4]) = v;
    }
    // Stage W transposed: ldsW[col][k] = W[k][col]; zero-pad cols >= n_cols
    for (int i = tid; i < DIN * DIN; i += 256) {
        int k = i >> 7, col = i & 127;   // consecutive tid -> consecutive col (coalesced)
        ldsW[col * XSTRIDE + k] = (col < n_cols) ? W[(size_t)k * ldW + col] : 0.0f;
    }
    __syncthreads();

    const int lane = tid & 31;
    const int wave = tid >> 5;
    const int m    = lane & 15;   // A row within tile / B,C col within tile
    const int hi   = lane >> 4;   // lane-half: K offset +2 (A,B) / M offset +8 (C)
    const int col  = wave * 16 + m;

    v8f acc0 = {}, acc1 = {}, acc2 = {}, acc3 = {};
    #pragma unroll 2
    for (int kk = 0; kk < DIN; kk += 4) {
        const int kb = kk + 2 * hi;
        v2f b  = *reinterpret_cast<const v2f*>(&ldsW[col * XSTRIDE + kb]);
        v2f a0 = *reinterpret_cast<const v2f*>(&ldsX[(m     ) * XSTRIDE + kb]);
        v2f a1 = *reinterpret_cast<const v2f*>(&ldsX[(m + 16) * XSTRIDE + kb]);
        v2f a2 = *reinterpret_cast<const v2f*>(&ldsX[(m + 32) * XSTRIDE + kb]);
        v2f a3 = *reinterpret_cast<const v2f*>(&ldsX[(m + 48) * XSTRIDE + kb]);
        acc0 = __builtin_amdgcn_wmma_f32_16x16x4_f32(false, a0, false, b, (short)0, acc0, false, false);
        acc1 = __builtin_amdgcn_wmma_f32_16x16x4_f32(false, a1, false, b, (short)0, acc1, false, false);
        acc2 = __builtin_amdgcn_wmma_f32_16x16x4_f32(false, a2, false, b, (short)0, acc2, false, false);
        acc3 = __builtin_amdgcn_wmma_f32_16x16x4_f32(false, a3, false, b, (short)0, acc3, false, false);
    }

    // Epilogue: C/D layout -> VGPR r holds M = r (lanes 0-15) / M = r+8 (lanes 16-31).
    // Base pointer computed once; all stores use compile-time immediate offsets.
    if (col < n_valid) {
        const int rbase = row0 + 8 * hi;
        float* p = Y + (size_t)rbase * LDY + col;
        if (row0 + ROWTILE <= n_rows) {       // uniform fast path (all but last block)
            #pragma unroll
            for (int r = 0; r < 8; ++r) p[(r +  0) * LDY] = acc0[r];
            #pragma unroll
            for (int r = 0; r < 8; ++r) p[(r + 16) * LDY] = acc1[r];
            #pragma unroll
            for (int r = 0; r < 8; ++r) p[(r + 32) * LDY] = acc2[r];
            #pragma unroll
            for (int r = 0; r < 8; ++r) p[(r + 48) * LDY] = acc3[r];
        } else {                              // guarded tail (last block only)
            #pragma unroll
            for (int r = 0; r < 8; ++r) if (rbase + r      < n_rows) p[(r +  0) * LDY] = acc0[r];
            #pragma unroll
            for (int r = 0; r < 8; ++r) if (rbase + r + 16 < n_rows) p[(r + 16) * LDY] = acc1[r];
            #pragma unroll
            for (int r = 0; r < 8; ++r) if (rbase + r + 32 < n_rows) p[(r + 32) * LDY] = acc2[r];
            #pragma unroll
            for (int r = 0; r < 8; ++r) if (rbase + r + 48 < n_rows) p[(r + 48) * LDY] = acc3[r];
        }
    }
}

// ---------------------------------------------------------------------------
// Edge scatter: agg[dst] += xw[src] * dinv[src]*dinv[dst]
// One wave32 per edge. F=128 fast path: float4 load + 4 atomics per lane.
// xw/agg are 51.2 MB -> L2-resident on MI455X (192 MB L2).
// ---------------------------------------------------------------------------
__global__ void __launch_bounds__(256)
scatter128(const float* __restrict__ xw, const int* __restrict__ src,
           const int* __restrict__ dst, const float* __restrict__ dinv,
           float* __restrict__ agg, int E) {
    int wid  = (blockIdx.x * blockDim.x + threadIdx.x) >> 5;
    int lane = threadIdx.x & 31;
    if (wid >= E) return;
    int s = src[wid], d = dst[wid];
    float coef = dinv[s] * dinv[d];
    float4 v = reinterpret_cast<const float4*>(xw + (size_t)s * DIN)[lane];
    float* ad = agg + (size_t)d * DIN + lane * 4;
    atomicAdd(ad + 0, v.x * coef);
    atomicAdd(ad + 1, v.y * coef);
    atomicAdd(ad + 2, v.z * coef);
    atomicAdd(ad + 3, v.w * coef);
}

__global__ void __launch_bounds__(256)
scatter_generic(const float* __restrict__ xw, const int* __restrict__ src,
                const int* __restrict__ dst, const float* __restrict__ dinv,
                float* __restrict__ agg, int E, int F, int ld) {
    int wid  = (blockIdx.x * blockDim.x + threadIdx.x) >> 5;
    int lane = threadIdx.x & 31;
    if (wid >= E) return;
    int s = src[wid], d = dst[wid];
    float coef = dinv[s] * dinv[d];
    for (int f = lane; f < F; f += 32)
        atomicAdd(&agg[(size_t)d * ld + f], xw[(size_t)s * ld + f] * coef);
}

// ---------------------------------------------------------------------------
// In-place self-loop + bias:  y += xw * dinv^2 + b, plus per-column partial
// sum / sum-of-squares for BN (one atomic pair per block per column).
// blockDim = 128, thread = one column, loops over a 128-row stripe.
// ---------------------------------------------------------------------------
__global__ void __launch_bounds__(128)
self_bias_stats(float* __restrict__ y, const float* __restrict__ xw,
                const float* __restrict__ dinv, const float* __restrict__ bias,
                float* __restrict__ stats, int n, int F, int ld,
                int rows_per_block, int do_stats) {
    int f = threadIdx.x;
    if (f >= F) return;
    int r0 = blockIdx.x * rows_per_block;
    int r1 = min(r0 + rows_per_block, n);
    float b = bias[f];
    float s = 0.0f, s2 = 0.0f;
    for (int r = r0; r < r1; ++r) {
        float di = dinv[r];
        size_t idx = (size_t)r * ld + f;
        float v = y[idx] + xw[idx] * (di * di) + b;
        y[idx] = v;
        s  += v;
        s2 += v * v;
    }
    if (do_stats) {
        atomicAdd(&stats[f],       s);
        atomicAdd(&stats[128 + f], s2);
    }
}

// stats[0..127]=sum, [128..255]=sumsq  ->  [256..383]=scale, [384..511]=shift
__global__ void finalize_stats(float* __restrict__ stats, const float* __restrict__ g,
                               const float* __restrict__ be, float invN) {
    int f = threadIdx.x;  // 128 threads
    float mu  = stats[f] * invN;
    float var = stats[128 + f] * invN - mu * mu;
    float a   = g[f] * rsqrtf(var + 1e-5f);
    stats[256 + f] = a;
    stats[384 + f] = be[f] - mu * a;
}

__global__ void __launch_bounds__(256)
bnrelu(const float* __restrict__ y, const float* __restrict__ stats,
       float* __restrict__ out, int n_elems) {
    int i = blockIdx.x * blockDim.x + threadIdx.x;
    if (i >= n_elems) return;
    int f = i & (DIN - 1);
    out[i] = fmaxf(y[i] * stats[256 + f] + stats[384 + f], 0.0f);
}

// ---------------------------------------------------------------------------
// log_softmax over 40 classes; one wave32 per row, __shfl_xor butterflies.
// Input stride 48 (padded), output stride 40.
// ---------------------------------------------------------------------------
__global__ void __launch_bounds__(256)
log_softmax40(const float* __restrict__ y, float* __restrict__ out, int n) {
    int row  = (blockIdx.x * blockDim.x + threadIdx.x) >> 5;
    int lane = threadIdx.x & 31;
    if (row >= n) return;
    const float* yr = y + (size_t)row * 48;
    float v0 = yr[lane];
    float v1 = (lane + 32 < 40) ? yr[lane + 32] : -INFINITY;
    float m = fmaxf(v0, v1);
    #pragma unroll
    for (int off = 16; off; off >>= 1) m = fmaxf(m, __shfl_xor(m, off, 32));
    float e = __expf(v0 - m) + ((lane + 32 < 40) ? __expf(v1 - m) : 0.0f);
    #pragma unroll
    for (int off = 16; off; off >>= 1) e += __shfl_xor(e, off, 32);
    float lse = m + __logf(e);
    out[(size_t)row * 40 + lane] = v0 - lse;
    if (lane + 32 < 40) out[(size_t)row * 40 + lane + 32] = v1 - lse;
}

// ---------------------------------------------------------------------------
extern "C" void kernel_launch(void* const* d_in, const int* in_sizes, int n_in,
                              void* d_out, int out_size, void* d_ws, size_t ws_size,
                              hipStream_t stream) {
    const float* x   = (const float*)d_in[0];
    const int*   ei  = (const int*)  d_in[1];
    const float* W1  = (const float*)d_in[2];
    const float* b1  = (const float*)d_in[3];
    const float* W2  = (const float*)d_in[4];
    const float* b2  = (const float*)d_in[5];
    const float* W3  = (const float*)d_in[6];
    const float* b3  = (const float*)d_in[7];
    const float* g1  = (const float*)d_in[8];
    const float* be1 = (const float*)d_in[9];
    const float* g2  = (const float*)d_in[10];
    const float* be2 = (const float*)d_in[11];

    const int N = in_sizes[0] / DIN;
    const int E = in_sizes[1] / 2;
    const int* src = ei;
    const int* dst = ei + E;
    float* out = (float*)d_out;

    // Workspace carving
    char* ws = (char*)d_ws;
    size_t off = 0;
    auto take = [&](size_t bytes) -> void* {
        void* p = ws + off;
        off += (bytes + 255) & ~(size_t)255;
        return p;
    };
    float* dinv  = (float*)take((size_t)N * 4);            // deg -> dinv in place
    float* A     = (float*)take((size_t)N * DIN * 4);      // xw
    float* B     = (float*)take((size_t)N * DIN * 4);      // agg / y
    float* C     = (float*)take((size_t)N * DIN * 4);      // h
    float* stats = (float*)take(512 * 4);
    (void)ws_size; (void)n_in; (void)out_size;

    const int gemm_blocks = (N + ROWTILE - 1) / ROWTILE;
    const int scat_blocks = (E * 32 + 255) / 256;   // one wave per edge
    const int sb_blocks   = (N + 127) / 128;
    const int ew_blocks   = (N * DIN + 255) / 256;

    // ---- degrees + normalization ----
    hipMemsetAsync(dinv, 0, (size_t)N * 4, stream);
    degree_count<<<(E + 255) / 256, 256, 0, stream>>>(dst, dinv, E);
    deg_to_dinv<<<(N + 255) / 256, 256, 0, stream>>>(dinv, N);

    // ---- Layer 1 ----
    gemm_wmma<DIN><<<gemm_blocks, 256, GEMM_LDS_BYTES, stream>>>(x, W1, A, N, DIN, DIN, DIN);
    hipMemsetAsync(B, 0, (size_t)N * DIN * 4, stream);
    hipMemsetAsync(stats, 0, 512 * 4, stream);
    scatter128<<<scat_blocks, 256, 0, stream>>>(A, src, dst, dinv, B, E);
    self_bias_stats<<<sb_blocks, 128, 0, stream>>>(B, A, dinv, b1, stats, N, DIN, DIN, 128, 1);
    finalize_stats<<<1, 128, 0, stream>>>(stats, g1, be1, 1.0f / (float)N);
    bnrelu<<<ew_blocks, 256, 0, stream>>>(B, stats, C, N * DIN);

    // ---- Layer 2 ----
    gemm_wmma<DIN><<<gemm_blocks, 256, GEMM_LDS_BYTES, stream>>>(C, W2, A, N, DIN, DIN, DIN);
    hipMemsetAsync(B, 0, (size_t)N * DIN * 4, stream);
    hipMemsetAsync(stats, 0, 512 * 4, stream);
    scatter128<<<scat_blocks, 256, 0, stream>>>(A, src, dst, dinv, B, E);
    self_bias_stats<<<sb_blocks, 128, 0, stream>>>(B, A, dinv, b2, stats, N, DIN, DIN, 128, 1);
    finalize_stats<<<1, 128, 0, stream>>>(stats, g2, be2, 1.0f / (float)N);
    bnrelu<<<ew_blocks, 256, 0, stream>>>(B, stats, C, N * DIN);

    // ---- Layer 3 (40 cols, padded stride 48) + log_softmax ----
    gemm_wmma<48><<<gemm_blocks, 256, GEMM_LDS_BYTES, stream>>>(C, W3, A, N, 40, 40, 40);
    hipMemsetAsync(B, 0, (size_t)N * 48 * 4, stream);
    scatter_generic<<<scat_blocks, 256, 0, stream>>>(A, src, dst, dinv, B, E, 40, 48);
    self_bias_stats<<<sb_blocks, 128, 0, stream>>>(B, A, dinv, b3, nullptr, N, 40, 48, 128, 0);
    log_softmax40<<<(N * 32 + 255) / 256, 256, 0, stream>>>(B, out, N);
}